// QueryAndGroup_12146167513657
// MI455X (gfx1250) — compile-verified
//
#include <hip/hip_runtime.h>

// Problem constants (from reference setup_inputs)
#define BATCH   4
#define NPTS    16384
#define MQ      2048
#define CF      64
#define NS      32
#define RADIUS2 0.04f
#define OUTCH   (3 + CF)
#define QPW     8          // queries per wave (rows 0..7 of the WMMA tile)

typedef __attribute__((ext_vector_type(2))) float v2f;
typedef __attribute__((ext_vector_type(8))) float v8f;

// ---------------------------------------------------------------------------
// Kernel 1: ball query via WMMA distance tiles.
// One wave handles 8 queries (tile rows 0..7; rows 8..15 are duplicates and
// never read back). Per iteration it computes an 8x32 d^2 sub-tile using two
// v_wmma_f32_16x16x4_f32 (d^2 = |q|^2 + |p|^2 - 2 q.p), then does an ordered
// ballot/popc append of in-radius point indices. The next 32-point tile is
// prefetched into registers before the selection so the L2 gather latency
// overlaps the SALU-heavy selection. Early-exits when all 8 queries are full.
// ---------------------------------------------------------------------------
__global__ __launch_bounds__(128) void ball_query_wmma_kernel(
    const float* __restrict__ xyz,      // (B, N, 3)
    const float* __restrict__ new_xyz,  // (B, M, 3)
    int* __restrict__ idx)              // (B*M, NS) in workspace
{
  const int lane  = threadIdx.x & 31;
  const int wave  = threadIdx.x >> 5;
  const int qbase = (blockIdx.x * 4 + wave) * QPW;   // 8 queries per wave
  const int b     = qbase / MQ;                      // uniform per wave (M%8==0)
  const int half  = lane >> 4;                       // K/lane-half select
  const int ml    = lane & 15;

  // ---- A matrix: 16x4, row m in lane ml, VGPR0/1 = K(2*half), K(2*half+1).
  // Rows 8..15 duplicate rows 0..7 (their D rows are ignored); never OOB.
  const float* q = new_xyz + (size_t)(qbase + (ml & 7)) * 3;
  const float qx = q[0], qy = q[1], qz = q[2];
  const float qsq = qx * qx + qy * qy + qz * qz;

  v2f a;
  a.x = -2.0f * (half ? qz : qx);   // fold the -2 of -2*q.p into A
  a.y = half ? 0.0f : (-2.0f * qy); // K=3 padded with zero

  // C/D row m = r + 8*half -> each lane seeds qsq of those rows (half=1 rows
  // are the duplicates; values only feed ignored rows).
  float qsqsel[8];
#pragma unroll
  for (int r = 0; r < 8; ++r)
    qsqsel[r] = __shfl(qsq, r + 8 * half, 32);

  int cnt[QPW], first[QPW];
#pragma unroll
  for (int m = 0; m < QPW; ++m) { cnt[m] = 0; first[m] = 0; }

  const float* xb = xyz + (size_t)b * NPTS * 3;

  // ---- software pipeline: preload tile for pbase = 0
  float p0x, p0y, p0z, p1x, p1y, p1z;
  {
    const float* p0 = xb + (size_t)ml * 3;
    const float* p1 = xb + (size_t)(16 + ml) * 3;
    p0x = p0[0]; p0y = p0[1]; p0z = p0[2];
    p1x = p1[0]; p1y = p1[1]; p1z = p1[2];
  }

  for (int pbase = 0; pbase < NPTS; pbase += 32) {
    // ---- prefetch next tile (clamped; values unused on the last iteration)
    const int nb = (pbase + 32 < NPTS) ? (pbase + 32) : 0;
    const float* n0 = xb + (size_t)(nb + ml) * 3;
    const float* n1 = xb + (size_t)(nb + 16 + ml) * 3;
    const float n0x = n0[0], n0y = n0[1], n0z = n0[2];
    const float n1x = n1[0], n1y = n1[1], n1z = n1[2];

    // ---- B matrices from current tile: 4x16, column n=ml in lane
    const float psq0 = p0x * p0x + p0y * p0y + p0z * p0z;
    const float psq1 = p1x * p1x + p1y * p1y + p1z * p1z;

    v2f b0, b1;
    b0.x = half ? p0z : p0x;  b0.y = half ? 0.0f : p0y;
    b1.x = half ? p1z : p1x;  b1.y = half ? 0.0f : p1y;

    v8f c0, c1;
#pragma unroll
    for (int r = 0; r < 8; ++r) {
      c0[r] = qsqsel[r] + psq0;   // |q|^2 + |p|^2 seeded into the accumulator
      c1[r] = qsqsel[r] + psq1;
    }

    // d^2 tiles: rows 0..7 valid, 16 points each
    v8f d0 = __builtin_amdgcn_wmma_f32_16x16x4_f32(
        false, a, false, b0, (short)0, c0, false, false);
    v8f d1 = __builtin_amdgcn_wmma_f32_16x16x4_f32(
        false, a, false, b1, (short)0, c1, false, false);

    bool alldone = true;
#pragma unroll
    for (int m = 0; m < QPW; ++m) {
      if (cnt[m] < NS) {                 // uniform: skip finished queries
        // row m lives in lanes 0..15 (m < 8)
        unsigned m0 = (unsigned)__ballot(d0[m] < RADIUS2) & 0xFFFFu;
        unsigned m1 = (unsigned)__ballot(d1[m] < RADIUS2) & 0xFFFFu;
        unsigned mask = m0 | (m1 << 16);   // bit n -> point pbase+n, ascending
        if (mask) {
          if (cnt[m] == 0) first[m] = pbase + __ffs(mask) - 1;
          int pos = cnt[m] + __popc(mask & ((1u << lane) - 1u));
          if (((mask >> lane) & 1u) && pos < NS)
            idx[(size_t)(qbase + m) * NS + pos] = pbase + lane;
          cnt[m] += __popc(mask);
        }
        alldone = alldone && (cnt[m] >= NS);
      }
    }
    if (alldone) break;   // uniform

    p0x = n0x; p0y = n0y; p0z = n0z;    // rotate pipeline
    p1x = n1x; p1y = n1y; p1z = n1z;
  }

  // Pad: slots >= cnt get first valid index (or 0 if none) — matches reference
#pragma unroll
  for (int m = 0; m < QPW; ++m) {
    if (cnt[m] < NS && lane >= cnt[m])
      idx[(size_t)(qbase + m) * NS + lane] = first[m];
  }
}

// ---------------------------------------------------------------------------
// Kernel 2: gather + concat. One wave per query; lane = sample.
// out layout (B, 67, M, NS): per channel a wave writes 32 consecutive floats
// (128B coalesced store); feature reads are 4B gathers that stay L2-resident.
// ---------------------------------------------------------------------------
__global__ __launch_bounds__(256) void group_gather_kernel(
    const float* __restrict__ xyz,       // (B, N, 3)
    const float* __restrict__ new_xyz,   // (B, M, 3)
    const float* __restrict__ features,  // (B, C, N)
    const int* __restrict__ idx,         // (B*M, NS)
    float* __restrict__ out)             // (B, 67, M, NS)
{
  const int lane = threadIdx.x & 31;
  const int wave = threadIdx.x >> 5;
  const int q    = blockIdx.x * 8 + wave;   // global query id
  const int b    = q / MQ;
  const int m    = q % MQ;

  const int i = idx[(size_t)q * NS + lane];

  const float* p  = xyz + ((size_t)b * NPTS + i) * 3;
  const float* nq = new_xyz + (size_t)q * 3;

  const size_t chstride = (size_t)MQ * NS;
  size_t o = (((size_t)b * OUTCH) * MQ + m) * NS + lane;

  out[o + 0 * chstride] = p[0] - nq[0];
  out[o + 1 * chstride] = p[1] - nq[1];
  out[o + 2 * chstride] = p[2] - nq[2];

  const float* fb = features + (size_t)b * CF * NPTS + i;
  o += 3 * chstride;
#pragma unroll 4
  for (int c = 0; c < CF; ++c) {
    out[o] = fb[(size_t)c * NPTS];
    o += chstride;
  }
}

extern "C" void kernel_launch(void* const* d_in, const int* in_sizes, int n_in,
                              void* d_out, int out_size, void* d_ws, size_t ws_size,
                              hipStream_t stream) {
  const float* xyz      = (const float*)d_in[0];  // (4,16384,3)
  const float* new_xyz  = (const float*)d_in[1];  // (4,2048,3)
  const float* features = (const float*)d_in[2];  // (4,64,16384)
  float* out = (float*)d_out;                     // (4,67,2048,32)
  int* idx = (int*)d_ws;                          // B*M*NS ints = 1 MB scratch

  // Kernel 1: 8 queries/wave, 4 waves/block -> 32 queries/block
  const int nq = BATCH * MQ;                      // 8192
  ball_query_wmma_kernel<<<nq / 32, 128, 0, stream>>>(xyz, new_xyz, idx);

  // Kernel 2: 1 query/wave, 8 waves/block
  group_gather_kernel<<<nq / 8, 256, 0, stream>>>(xyz, new_xyz, features, idx, out);
}